// myGIN_48009144435167
// MI455X (gfx1250) — compile-verified
//
#include <hip/hip_runtime.h>

typedef float v2f __attribute__((ext_vector_type(2)));
typedef float v8f __attribute__((ext_vector_type(8)));

#define D 128
#define GIN_BN_EPS 1e-5f
#define GEMM_ROWS 80   // rows per block = 5 tiles of 16 (100000 = 1250 * 80)
#define GEMM_TILES 5   // row tiles per wave (independent WMMA chains)

// -------------------- zero fill --------------------
__global__ __launch_bounds__(256) void k_zero(float* __restrict__ p, long long n) {
  long long i = (long long)blockIdx.x * blockDim.x + threadIdx.x;
  const long long stride = (long long)gridDim.x * blockDim.x;
  for (; i < n; i += stride) p[i] = 0.0f;
}

// -------------------- scatter-add aggregation --------------------
// One wave per edge; lane L moves floats [4L, 4L+4) of the 128-wide row.
// src/dst loads are wave-uniform; feature traffic is fully coalesced float4.
// x (51MB), agg (51MB), edges (12.8MB) are L2-resident (192MB): this phase is
// L2 atomic-throughput bound, which is the best this access pattern allows.
__global__ __launch_bounds__(256) void k_scatter(const float* __restrict__ x,
                                                 const int* __restrict__ src,
                                                 const int* __restrict__ dst,
                                                 float* __restrict__ agg,
                                                 long long E) {
  const long long t = (long long)blockIdx.x * blockDim.x + threadIdx.x;
  const long long e = t >> 5;
  if (e >= E) return;
  const int c = (int)(t & 31) << 2;
  const long long s = src[e];
  const long long d = dst[e];
  const float4 v = *(const float4*)(x + s * D + c);
  float* p = agg + d * D + c;
  atomicAdd(p + 0, v.x);
  atomicAdd(p + 1, v.y);
  atomicAdd(p + 2, v.z);
  atomicAdd(p + 3, v.w);
}

// -------------------- fused (A [+ Aadd]) @ W + bias, ReLU --------------------
// Block: 256 threads = 8 waves; block computes an 80-row strip x 128 cols.
// Wave w owns cols [16w, 16w+16): it builds its 128x16 B slab once in 64 VGPRs
// and reuses it for 5 row tiles (5 independent v8f accumulator chains), so the
// column-strided weight fetches are amortized 5x and the XDL pipe always has
// an independent WMMA to issue. V_WMMA_F32_16X16X4_F32 keeps full fp32.
// 32-bit operand striping (ISA 7.12.2): lanes 0-15 hold K=k0,k0+1; lanes
// 16-31 hold K=k0+2,k0+3 (A rows / B cols indexed by lane&15).
__global__ __launch_bounds__(256) void k_gemm_bias_relu(
    const float* __restrict__ A, const float* __restrict__ Aadd,
    const float* __restrict__ W, const float* __restrict__ bias,
    float* __restrict__ out, int N) {
  __shared__ float sA[GEMM_ROWS][D];  // 40 KB staged A strip (fused x+agg add)

  const int tid = threadIdx.x;
  const int wave = tid >> 5;
  const int lane = tid & 31;
  const long long rowBase = (long long)blockIdx.x * GEMM_ROWS;

  // Stage A (+ optional elementwise add) into LDS: 10240 floats, 40/thread.
  for (int i = tid * 4; i < GEMM_ROWS * D; i += 256 * 4) {
    const int r = i >> 7;
    const int c = i & (D - 1);
    long long row = rowBase + r;
    if (row >= N) row = N - 1;  // clamp (harmless duplicate stage)
    float4 a = *(const float4*)(A + row * D + c);
    if (Aadd) {
      const float4 g = *(const float4*)(Aadd + row * D + c);
      a.x += g.x; a.y += g.y; a.z += g.z; a.w += g.w;
    }
    *(float4*)&sA[r][c] = a;
  }
  __syncthreads();

  const int n0 = wave * 16;
  const int mlane = lane & 15;
  const int khalf = (lane >> 4) * 2;  // 0 for lanes 0-15, 2 for lanes 16-31
  const int ncol = n0 + mlane;

  // Preload this wave's 128x16 B slab: per k-step lane holds W[kk][n], W[kk+1][n].
  v2f bfrag[32];
#pragma unroll
  for (int ks = 0; ks < 32; ++ks) {
    const int kk = ks * 4 + khalf;
    bfrag[ks].x = W[kk * D + ncol];
    bfrag[ks].y = W[(kk + 1) * D + ncol];
  }

  v8f acc[GEMM_TILES];
#pragma unroll
  for (int t = 0; t < GEMM_TILES; ++t)
    acc[t] = (v8f){0.f, 0.f, 0.f, 0.f, 0.f, 0.f, 0.f, 0.f};

#pragma unroll
  for (int ks = 0; ks < 32; ++ks) {
    const int kk = ks * 4 + khalf;
#pragma unroll
    for (int t = 0; t < GEMM_TILES; ++t) {
      const v2f afrag = *(const v2f*)&sA[t * 16 + mlane][kk];  // kk even: 8B aligned
      acc[t] = __builtin_amdgcn_wmma_f32_16x16x4_f32(
          false, afrag, false, bfrag[ks], (short)0, acc[t], false, false);
    }
  }

  // D layout: VGPR r -> M=r (lanes 0-15) / M=r+8 (lanes 16-31), N = lane&15.
  const float bcol = bias[ncol];
#pragma unroll
  for (int t = 0; t < GEMM_TILES; ++t) {
#pragma unroll
    for (int r = 0; r < 8; ++r) {
      const int m = t * 16 + ((lane < 16) ? r : (r + 8));
      const long long row = rowBase + m;
      if (row < N) {
        float v = acc[t][r] + bcol;
        v = v > 0.f ? v : 0.f;  // ReLU
        out[row * D + ncol] = v;
      }
    }
  }
}

// -------------------- BN column statistics (sum / sumsq) --------------------
__global__ __launch_bounds__(256) void k_bn_stats(const float* __restrict__ v,
                                                  float* __restrict__ stats,
                                                  int N) {
  __shared__ float sh[256];
  const int col = threadIdx.x & (D - 1);
  const int half = threadIdx.x >> 7;  // two rows per block-iteration
  float s = 0.f, s2 = 0.f;
  for (int r = blockIdx.x * 2 + half; r < N; r += gridDim.x * 2) {
    const float val = v[(long long)r * D + col];
    s += val;
    s2 = fmaf(val, val, s2);
  }
  sh[threadIdx.x] = s;
  __syncthreads();
  if (half == 0) atomicAdd(&stats[col], s + sh[threadIdx.x + 128]);
  __syncthreads();
  sh[threadIdx.x] = s2;
  __syncthreads();
  if (half == 0) atomicAdd(&stats[D + col], s2 + sh[threadIdx.x + 128]);
}

// -------------------- BN apply: (v-mean)*rsqrt(var+eps)*g + b --------------------
__global__ __launch_bounds__(256) void k_bn_apply(
    const float* __restrict__ v, const float* __restrict__ stats,
    const float* __restrict__ gamma, const float* __restrict__ beta,
    float* __restrict__ out, long long nquads, float invN) {
  const long long q = (long long)blockIdx.x * blockDim.x + threadIdx.x;
  if (q >= nquads) return;
  const long long i = q * 4;
  const int c = (int)(i & (D - 1));
  const float4 val = *(const float4*)(v + i);
  const float vin[4] = {val.x, val.y, val.z, val.w};
  float vo[4];
#pragma unroll
  for (int j = 0; j < 4; ++j) {
    const int cc = c + j;
    const float mean = stats[cc] * invN;
    const float var = stats[D + cc] * invN - mean * mean;  // biased (jnp.var)
    vo[j] = (vin[j] - mean) * rsqrtf(var + GIN_BN_EPS) * gamma[cc] + beta[cc];
  }
  const float4 o = {vo[0], vo[1], vo[2], vo[3]};
  *(float4*)(out + i) = o;
}

extern "C" void kernel_launch(void* const* d_in, const int* in_sizes, int n_in,
                              void* d_out, int out_size, void* d_ws, size_t ws_size,
                              hipStream_t stream) {
  (void)n_in; (void)out_size; (void)ws_size;

  const float* x   = (const float*)d_in[0];
  const int* edges = (const int*)d_in[1];
  const float* W1a = (const float*)d_in[2];
  const float* b1a = (const float*)d_in[3];
  const float* W1b = (const float*)d_in[4];
  const float* b1b = (const float*)d_in[5];
  const float* g1  = (const float*)d_in[6];
  const float* be1 = (const float*)d_in[7];
  const float* W2a = (const float*)d_in[8];
  const float* b2a = (const float*)d_in[9];
  const float* W2b = (const float*)d_in[10];
  const float* b2b = (const float*)d_in[11];
  const float* g2  = (const float*)d_in[12];
  const float* be2 = (const float*)d_in[13];

  const int N = in_sizes[0] / D;        // 100000
  const long long E = in_sizes[1] / 2;  // 1.6M
  const int* src = edges;
  const int* dst = edges + E;

  const long long ND = (long long)N * D;
  float* agg   = (float*)d_ws;   // also reused as v (post-MLP, pre-BN)
  float* u     = agg + ND;       // MLP hidden
  float* h1    = u + ND;         // layer-1 output
  float* stats = h1 + ND;        // 256 floats: [sum(128) | sumsq(128)]

  float* outp = (float*)d_out;
  const float invN = 1.0f / (float)N;

  const dim3 blk(256);
  const int zeroBlocks    = 2048;
  const int scatterBlocks = (int)((E * 32 + 255) / 256);
  const int gemmBlocks    = (N + GEMM_ROWS - 1) / GEMM_ROWS;  // 1250
  const int bnStatBlocks  = 512;
  const long long nquads  = ND / 4;
  const int bnApplyBlocks = (int)((nquads + 255) / 256);

  for (int layer = 0; layer < 2; ++layer) {
    const float* xin = (layer == 0) ? x : h1;
    const float* Wa  = (layer == 0) ? W1a : W2a;
    const float* ba  = (layer == 0) ? b1a : b2a;
    const float* Wb  = (layer == 0) ? W1b : W2b;
    const float* bb  = (layer == 0) ? b1b : b2b;
    const float* g   = (layer == 0) ? g1  : g2;
    const float* be  = (layer == 0) ? be1 : be2;
    float* xout      = (layer == 0) ? h1  : outp;

    k_zero<<<zeroBlocks, blk, 0, stream>>>(agg, ND);
    k_zero<<<1, blk, 0, stream>>>(stats, 256);
    k_scatter<<<scatterBlocks, blk, 0, stream>>>(xin, src, dst, agg, E);
    // u = relu((xin + agg) @ Wa + ba)
    k_gemm_bias_relu<<<gemmBlocks, blk, 0, stream>>>(xin, agg, Wa, ba, u, N);
    // v = relu(u @ Wb + bb)   (v reuses agg's buffer; agg already consumed)
    k_gemm_bias_relu<<<gemmBlocks, blk, 0, stream>>>(u, nullptr, Wb, bb, agg, N);
    k_bn_stats<<<bnStatBlocks, blk, 0, stream>>>(agg, stats, N);
    k_bn_apply<<<bnApplyBlocks, blk, 0, stream>>>(agg, stats, g, be, xout,
                                                  nquads, invN);
  }
}